// RetrosynthesisAttention_29351806501170
// MI455X (gfx1250) — compile-verified
//
#include <hip/hip_runtime.h>
#include <cmath>

typedef __attribute__((ext_vector_type(2))) float v2f;
typedef __attribute__((ext_vector_type(8))) float v8f;

#define B_  16
#define N_  256
#define M_  256
#define D_  256
#define U_  256

// ---------------------------------------------------------------------------
// Native gfx1250 tanh (V_TANH_F32). Falls back to inline asm if the builtin
// is unavailable in this toolchain.
// ---------------------------------------------------------------------------
__device__ __forceinline__ float fast_tanh(float x) {
#if defined(__has_builtin)
#if __has_builtin(__builtin_amdgcn_tanhf)
    return __builtin_amdgcn_tanhf(x);
#else
    float r;
    asm volatile("v_tanh_f32 %0, %1" : "=v"(r) : "v"(x));
    return r;
#endif
#else
    float r;
    asm volatile("v_tanh_f32 %0, %1" : "=v"(r) : "v"(x));
    return r;
#endif
}

__device__ __forceinline__ float fast_exp(float x) {
    // exp(x) = exp2(x * log2(e)) -> v_exp_f32
    return __builtin_amdgcn_exp2f(x * 1.44269504088896340736f);
}

// ---------------------------------------------------------------------------
// Wave-level 16x16 f32 GEMM tile using V_WMMA_F32_16X16X4_F32.
// A: 16xK row-major (lda), B: KxN row-major (ldb), C: 16x16 tile (ldc).
// A-matrix VGPR layout (32-bit, 16x4): lanes 0-15 -> M=lane, K={k,k+1};
// lanes 16-31 -> M=lane-16, K={k+2,k+3}.
// B-matrix (4x16): half-wave split covers K halves.
// C/D: VGPR r -> (M=r, N=lane) lanes 0-15; (M=r+8, N=lane-16) lanes 16-31.
// ---------------------------------------------------------------------------
__device__ __forceinline__ void wmma_tile_f32(const float* __restrict__ A, int lda,
                                              const float* __restrict__ Bm, int ldb,
                                              float* __restrict__ C, int ldc,
                                              const float* __restrict__ bias,
                                              int K) {
    const int lane = threadIdx.x & 31;
    const int half = lane >> 4;     // 0: K pair {0,1}, 1: K pair {2,3}
    const int l16  = lane & 15;

    v8f c = {};
    for (int k = 0; k < K; k += 4) {
        const int kk = k + 2 * half;
        v2f a, b;
        a.x = A[l16 * lda + kk];
        a.y = A[l16 * lda + kk + 1];
        b.x = Bm[kk * ldb + l16];
        b.y = Bm[(kk + 1) * ldb + l16];
        c = __builtin_amdgcn_wmma_f32_16x16x4_f32(
                /*neg_a=*/false, a, /*neg_b=*/false, b,
                /*c_mod=*/(short)0, c, /*reuse_a=*/false, /*reuse_b=*/false);
    }
    const float bv = bias ? bias[l16] : 0.0f;
#pragma unroll
    for (int r = 0; r < 8; ++r) {
        const int row = r + 8 * half;
        C[row * ldc + l16] = c[r] + bv;
    }
}

// ---------------------------------------------------------------------------
// Projection GEMM: Y[rows,U] = X[rows,D] @ W[D,U] + bias.   rows = 4096.
// 4 waves per block, one 16x16 tile per wave. 4096 tiles total.
// ---------------------------------------------------------------------------
__global__ void proj_gemm(const float* __restrict__ X,
                          const float* __restrict__ W,
                          const float* __restrict__ bias,
                          float* __restrict__ Y) {
    const int t  = blockIdx.x * 4 + (threadIdx.x >> 5);   // tile id
    const int tr = t >> 4;          // row tile (0..255)
    const int tc = t & 15;          // col tile (0..15)
    wmma_tile_f32(X + tr * 16 * D_, D_,
                  W + tc * 16, U_,
                  Y + tr * 16 * U_ + tc * 16, U_,
                  bias + tc * 16, D_);
}

// ---------------------------------------------------------------------------
// Fused logits + softmax.
// Block = 256 threads, handles one batch b and 8 decoder positions n0..n0+7.
// logit[b,n,m] = v_b + sum_u tanh(enc_proj[b,m,u] + dec_proj[b,n,u]) * v_w[u]
// Then softmax over m (the 256 threads of the block), weights -> global.
// Each 16B enc_proj load is amortized over 8 n values -> 32 tanh+FMA per load.
// ---------------------------------------------------------------------------
__global__ void attn_logits_softmax(const float* __restrict__ enc_proj,
                                    const float* __restrict__ dec_proj,
                                    const float* __restrict__ v_w,
                                    const float* __restrict__ v_b,
                                    float* __restrict__ weights_out) {
    constexpr int NPB = 8;                    // n per block
    __shared__ float4 dps[NPB][U_ / 4];       // dec_proj rows
    __shared__ float4 vws[U_ / 4];            // v_w
    __shared__ float  red[M_];                // reduction scratch

    const int b   = blockIdx.x >> 5;          // 32 n-tiles per batch
    const int n0  = (blockIdx.x & 31) * NPB;
    const int tid = threadIdx.x;              // m index

#pragma unroll
    for (int j = 0; j < NPB; ++j)
        ((float*)dps[j])[tid] = dec_proj[((size_t)(b * N_ + n0 + j)) * U_ + tid];
    ((float*)vws)[tid] = v_w[tid];
    __syncthreads();

    const float vb = v_b[0];
    float acc[NPB];
#pragma unroll
    for (int j = 0; j < NPB; ++j) acc[j] = vb;

    const float4* ep4 = (const float4*)(enc_proj + (size_t)(b * M_ + tid) * U_);
    for (int u4 = 0; u4 < U_ / 4; ++u4) {
        const float4 e = ep4[u4];
        const float4 w = vws[u4];
#pragma unroll
        for (int j = 0; j < NPB; ++j) {
            const float4 d = dps[j][u4];
            acc[j] += fast_tanh(e.x + d.x) * w.x;
            acc[j] += fast_tanh(e.y + d.y) * w.y;
            acc[j] += fast_tanh(e.z + d.z) * w.z;
            acc[j] += fast_tanh(e.w + d.w) * w.w;
        }
    }

    // Softmax over m (= over the 256 threads) for each of the 8 n values.
    for (int j = 0; j < NPB; ++j) {
        __syncthreads();
        red[tid] = acc[j];
        __syncthreads();
        for (int s = 128; s > 0; s >>= 1) {
            if (tid < s) red[tid] = fmaxf(red[tid], red[tid + s]);
            __syncthreads();
        }
        const float mx = red[0];
        __syncthreads();
        const float e = fast_exp(acc[j] - mx);
        red[tid] = e;
        __syncthreads();
        for (int s = 128; s > 0; s >>= 1) {
            if (tid < s) red[tid] += red[tid + s];
            __syncthreads();
        }
        const float wgt = e / red[0];
        weights_out[((size_t)(b * N_ + n0 + j)) * M_ + tid] = wgt;
    }
}

// ---------------------------------------------------------------------------
// Context GEMM (batched): ctx[b] (NxD) = weights[b] (NxM) @ enc[b] (MxD).
// 4 waves per block, one 16x16 tile per wave; 256 tiles per batch.
// ---------------------------------------------------------------------------
__global__ void context_gemm(const float* __restrict__ Wt,
                             const float* __restrict__ enc,
                             float* __restrict__ ctx) {
    const int t    = blockIdx.x * 4 + (threadIdx.x >> 5);
    const int bt   = t >> 8;                  // batch (256 tiles each)
    const int tile = t & 255;
    const int tr   = tile >> 4;               // n tile
    const int tc   = tile & 15;               // d tile
    const size_t boff = (size_t)bt * N_ * M_;
    wmma_tile_f32(Wt + boff + tr * 16 * M_, M_,
                  enc + boff + tc * 16, D_,
                  ctx + boff + tr * 16 * D_ + tc * 16, D_,
                  nullptr, M_);
}

// ---------------------------------------------------------------------------
extern "C" void kernel_launch(void* const* d_in, const int* in_sizes, int n_in,
                              void* d_out, int out_size, void* d_ws, size_t ws_size,
                              hipStream_t stream) {
    const float* enc = (const float*)d_in[0];   // [B,M,D]
    const float* dec = (const float*)d_in[1];   // [B,N,D]
    const float* W1  = (const float*)d_in[2];   // [D,U]
    const float* b1  = (const float*)d_in[3];   // [U]
    const float* W2  = (const float*)d_in[4];   // [D,U]
    const float* b2  = (const float*)d_in[5];   // [U]
    const float* vw  = (const float*)d_in[6];   // [U,1]
    const float* vb  = (const float*)d_in[7];   // [1]

    float* ctx = (float*)d_out;                         // [B,N,D]
    float* wts = (float*)d_out + (size_t)B_ * N_ * D_;  // [B,N,M,1]

    float* enc_proj = (float*)d_ws;                         // [B,M,U] 4MB
    float* dec_proj = enc_proj + (size_t)B_ * M_ * U_;      // [B,N,U] 4MB

    // 256 row-tiles x 16 col-tiles = 4096 tiles; 4 waves (tiles) per block.
    proj_gemm<<<1024, 128, 0, stream>>>(enc, W1, b1, enc_proj);
    proj_gemm<<<1024, 128, 0, stream>>>(dec, W2, b2, dec_proj);

    // 16 batches * 32 n-tiles = 512 blocks of 256 threads.
    attn_logits_softmax<<<512, 256, 0, stream>>>(enc_proj, dec_proj, vw, vb, wts);

    // 16 batches * 256 tiles = 4096 tiles; 4 waves per block.
    context_gemm<<<1024, 128, 0, stream>>>(wts, enc, ctx);
}